// CorePartLayer_807453852264
// MI455X (gfx1250) — compile-verified
//
#include <hip/hip_runtime.h>
#include <hip/hip_bf16.h>

// ---------------------------------------------------------------------------
// Problem constants (from reference):
//   z  : (32, 64) f32          L : (64,) f32
//   U  : (64, 262144) f32      mu: (262144,) f32
//   proj = (L*z) @ U + mu  -> (32, 262144) -> reshape (32,8,32,32,32)
//   out  = zeros(32,8,64,64,64); out[:,:,16:48,16:48,16:48] = block
//
// Roofline: 1.07 GFLOP vs ~323 MB minimum traffic (67 MB U read, 224 MB zero
// periphery + 32 MB core stores) -> purely HBM-bound (~14 us @ 23.3 TB/s).
// Strategy: f32 WMMA (16x16x4) GEMM writing the core directly; separate
// kernel zeroes only the periphery; NT hints on all streaming traffic since
// none of it is re-read (keeps the 192 MB L2 clean).
// ---------------------------------------------------------------------------
#define NB     64
#define BATCH  32
#define DIMN   262144        // 8*32*32*32
#define OUT_PER_B 2097152    // 8*64*64*64
#define POS    16
#define LZ_STRIDE 66         // 64 + 2 pad -> conflict-free strided LDS reads

typedef __attribute__((ext_vector_type(2))) float v2f;
typedef __attribute__((ext_vector_type(4))) float v4f;
typedef __attribute__((ext_vector_type(8))) float v8f;

// ---------------------------------------------------------------------------
// Kernel 1: zero the periphery only (224 MB of the 256 MB output).
// 4-float chunks; a chunk is fully inside the core iff i,j in [16,48) and
// chunk index kk in [4,12) (i.e. k in [16,48)). Those are written by kernel 2.
// ---------------------------------------------------------------------------
__global__ void __launch_bounds__(256)
zero_periphery_kernel(float* __restrict__ out)
{
    unsigned q = blockIdx.x * 256u + threadIdx.x;   // chunk id, 4 floats/chunk
    unsigned kk = q & 15u;            // 16 chunks per 64-float k-row
    unsigned j  = (q >> 4) & 63u;
    unsigned i  = (q >> 10) & 63u;
    bool core = (i - 16u < 32u) & (j - 16u < 32u) & (kk - 4u < 8u);
    if (!core) {
        v4f zero = {0.0f, 0.0f, 0.0f, 0.0f};
        // Streaming store, never re-read: non-temporal.
        __builtin_nontemporal_store(zero, (v4f*)(out + (size_t)q * 4u));
    }
}

// ---------------------------------------------------------------------------
// Kernel 2: WMMA GEMM, writing results directly into the core region.
// One wave per 16-column tile of DIM; each wave computes all 32 batch rows
// (two 16x16x4 f32 WMMA chains sharing B fragments -> 2x reuse of U bytes).
// ---------------------------------------------------------------------------
__global__ void __launch_bounds__(256)
gemm_core_kernel(const float* __restrict__ z,
                 const float* __restrict__ U,
                 const float* __restrict__ L,
                 const float* __restrict__ mu,
                 float* __restrict__ out)
{
    __shared__ float lz[BATCH * LZ_STRIDE];   // scaled z: lz[m][k] = L[k]*z[m][k]

    const int tid = threadIdx.x;

    // Stage L*z into LDS (32*64 = 2048 elements, 8 per thread).
#pragma unroll
    for (int s = 0; s < 8; ++s) {
        int idx = s * 256 + tid;
        int m = idx >> 6;
        int k = idx & 63;
        lz[m * LZ_STRIDE + k] = L[k] * z[m * NB + k];
    }
    __syncthreads();

    const int wave = tid >> 5;
    const int lane = tid & 31;
    const int half = lane >> 4;          // 0: K+{0,1}/M=lane, 1: K+{2,3}/M=lane-16
    const int l15  = lane & 15;

    const int t  = blockIdx.x * 8 + wave;        // tile id 0..16383
    const long n0 = (long)t * 16;                // first column of tile

    // C init = mu[n] broadcast down the M (batch) dimension of the tile.
    const float muv = mu[n0 + l15];
    v8f acc_lo, acc_hi;
#pragma unroll
    for (int r = 0; r < 8; ++r) { acc_lo[r] = muv; acc_hi[r] = muv; }

    const float* ucol = U + n0 + l15;            // this lane's U column
    const float* alo  = &lz[l15 * LZ_STRIDE];          // batches 0..15
    const float* ahi  = &lz[(l15 + 16) * LZ_STRIDE];   // batches 16..31

#pragma unroll
    for (int kk = 0; kk < 16; ++kk) {
        const int krow = kk * 4 + half * 2;      // ISA 16x16x4 f32 K striping

        // B fragment (4x16): two coalesced 64B row segments per half-wave.
        // U is read exactly once across the whole kernel: non-temporal.
        v2f bf;
        bf[0] = __builtin_nontemporal_load(ucol + (long)krow * DIMN);
        bf[1] = __builtin_nontemporal_load(ucol + (long)(krow + 1) * DIMN);

        // A fragments from LDS (ds_load_b64, conflict-free via row padding).
        v2f af_lo = *(const v2f*)(alo + krow);
        v2f af_hi = *(const v2f*)(ahi + krow);

        acc_lo = __builtin_amdgcn_wmma_f32_16x16x4_f32(
            false, af_lo, false, bf, (short)0, acc_lo, false, false);
        acc_hi = __builtin_amdgcn_wmma_f32_16x16x4_f32(
            false, af_hi, false, bf, (short)0, acc_hi, false, false);
    }

    // Decompose tile -> (c,i,j,k0) in the 32^3 core, map to 64^3 grid at +16.
    const int k0 = (t & 1) * 16;
    const int j  = (t >> 1) & 31;
    const int i  = (t >> 6) & 31;
    const int c  =  t >> 11;
    const long base = (long)c * (64 * 64 * 64)
                    + (long)(i + POS) * (64 * 64)
                    + (long)(j + POS) * 64
                    + (k0 + POS) + l15;

    // D layout: VGPR r -> M=r (lanes 0-15) / M=r+8 (lanes 16-31).
    // Output is write-once streaming: non-temporal stores.
#pragma unroll
    for (int r = 0; r < 8; ++r) {
        const int b = r + half * 8;
        __builtin_nontemporal_store(acc_lo[r], out + (long)b        * OUT_PER_B + base);
        __builtin_nontemporal_store(acc_hi[r], out + (long)(b + 16) * OUT_PER_B + base);
    }
}

// ---------------------------------------------------------------------------
extern "C" void kernel_launch(void* const* d_in, const int* in_sizes, int n_in,
                              void* d_out, int out_size, void* d_ws, size_t ws_size,
                              hipStream_t stream)
{
    const float* z  = (const float*)d_in[0];
    const float* U  = (const float*)d_in[1];
    const float* L  = (const float*)d_in[2];
    const float* mu = (const float*)d_in[3];
    float* out = (float*)d_out;

    // 67,108,864 output floats -> 16,777,216 four-float chunks.
    zero_periphery_kernel<<<65536, 256, 0, stream>>>(out);

    // 16384 column tiles, 8 waves (tiles) per 256-thread block.
    gemm_core_kernel<<<2048, 256, 0, stream>>>(z, U, L, mu, out);
}